// EventSampler_82429012345542
// MI455X (gfx1250) — compile-verified
//
#include <hip/hip_runtime.h>
#include <hip/hip_bf16.h>
#include <math.h>

#define BB 8
#define NN 16384
#define DD 256
#define PP 3
#define NUM_SAMPLES 500
#define MIN_K 10
#define CAND_CAP 1024

typedef __attribute__((ext_vector_type(2))) float v2f;
typedef __attribute__((ext_vector_type(8))) float v8f;

__device__ __forceinline__ unsigned fkey(float f) {
  unsigned u = __float_as_uint(f);
  return (u & 0x80000000u) ? ~u : (u | 0x80000000u);
}

// ---------------- Kernel 1: valid counts -> k, zero candidate counters ----
__global__ void k_count(const int* __restrict__ mask, int* __restrict__ kvals,
                        unsigned* __restrict__ counters) {
  __shared__ int red[256];
  const int b = blockIdx.x;
  const int t = threadIdx.x;
  int s = 0;
  for (int i = t; i < NN; i += 256) s += mask[b * NN + i];
  red[t] = s;
  __syncthreads();
  for (int off = 128; off > 0; off >>= 1) {
    if (t < off) red[t] += red[t + off];
    __syncthreads();
  }
  if (t == 0) {
    int v = red[0];
    int k = v < MIN_K ? MIN_K : (v > NUM_SAMPLES ? NUM_SAMPLES : v);
    kvals[b] = k;
    counters[b] = 0u;
  }
}

// ---------------- Kernel 2: scores via V_WMMA_F32_16X16X4_F32 -------------
// One wave handles a 16-row tile; B-matrix columns all hold W so every
// output column is the score vector.  f32 WMMA keeps exact f32 accumulate.
__global__ void k_scores(const float* __restrict__ feat,
                         const float* __restrict__ W,
                         const float* __restrict__ bias_p,
                         const int* __restrict__ mask,
                         float* __restrict__ scores) {
  const int lane = threadIdx.x & 31;
  const int wave = threadIdx.x >> 5;
  const int tile = blockIdx.x * 8 + wave;   // tiles of 16 rows over B*N
  const int row0 = tile * 16;
  const int m    = lane & 15;
  const int koff = (lane >> 4) * 2;         // lanes 0-15: K 0,1  lanes 16-31: K 2,3
  const float bias = bias_p[0];

  const float* __restrict__ arow = feat + (size_t)(row0 + m) * DD + koff;
  const float* __restrict__ wrow = W + koff;

  __builtin_prefetch(arow, 0, 0);           // global_prefetch_b8

  v8f c = {0.f, 0.f, 0.f, 0.f, 0.f, 0.f, 0.f, 0.f};
#pragma unroll 8
  for (int k = 0; k < DD; k += 4) {
    v2f a  = *(const v2f*)(arow + k);
    v2f w2 = *(const v2f*)(wrow + k);
    c = __builtin_amdgcn_wmma_f32_16x16x4_f32(false, a, false, w2,
                                              (short)0, c, false, false);
  }
  // C/D layout: VGPR r holds M=r (lanes 0-15) or M=r+8 (lanes 16-31), N=lane%16.
  // Column 0 (lane 0 / lane 16) carries the 16 scores.
  if ((lane & 15) == 0) {
    const int base = row0 + (lane >> 4) * 8;
#pragma unroll
    for (int r = 0; r < 8; ++r) {
      float s = c[r] + bias;
      const int gi = base + r;
      scores[gi] = mask[gi] ? s : -__builtin_inff();
    }
  }
}

// ---------------- Kernel 3: per-batch radix select (k-th largest key) -----
__global__ void k_radix(const float* __restrict__ scores,
                        const int* __restrict__ kvals,
                        unsigned* __restrict__ Tout) {
  __shared__ unsigned hist[256];
  __shared__ unsigned s_prefix;
  __shared__ int s_want;
  const int b = blockIdx.x;
  const int t = threadIdx.x;
  if (t == 0) { s_prefix = 0u; s_want = kvals[b]; }
  __syncthreads();
  for (int pass = 3; pass >= 0; --pass) {
    hist[t] = 0u;
    __syncthreads();
    const unsigned prefix = s_prefix;
    const int shift = pass * 8;
    for (int i = t; i < NN; i += 256) {
      const unsigned key = fkey(scores[b * NN + i]);
      const unsigned hi = (pass == 3) ? 0u : (key >> (shift + 8));
      if (hi == prefix) atomicAdd(&hist[(key >> shift) & 255u], 1u);
    }
    __syncthreads();
    if (t == 0) {
      int want = s_want;
      unsigned cum = 0u;
      for (int bin = 255; bin >= 0; --bin) {
        const unsigned nc = cum + hist[bin];
        if ((int)nc >= want) {
          s_want = want - (int)cum;
          s_prefix = (s_prefix << 8) | (unsigned)bin;
          break;
        }
        cum = nc;
      }
    }
    __syncthreads();
  }
  if (t == 0) Tout[b] = s_prefix;
}

// ---------------- Kernel 4: compact candidates (key >= threshold) ---------
__global__ void k_compact(const float* __restrict__ scores,
                          const unsigned* __restrict__ Tout,
                          unsigned* __restrict__ counters,
                          unsigned long long* __restrict__ cand) {
  const int b = blockIdx.y;
  const unsigned T = Tout[b];
  const int i = blockIdx.x * blockDim.x + threadIdx.x;
  if (i < NN) {
    const unsigned key = fkey(scores[b * NN + i]);
    if (key >= T) {
      const unsigned pos = atomicAdd(&counters[b], 1u);
      if (pos < CAND_CAP)
        cand[b * CAND_CAP + pos] =
            ((unsigned long long)key << 32) | (unsigned)(~(unsigned)i);
    }
  }
}

// ---------------- Kernel 5: bitonic sort 1024 candidates, emit top-500 ----
__global__ void k_sort(const unsigned long long* __restrict__ cand,
                       const unsigned* __restrict__ counters,
                       int* __restrict__ tk_idx, float* __restrict__ tk_sc) {
  __shared__ unsigned long long s[CAND_CAP];
  const int b = blockIdx.x;
  const int t = threadIdx.x;  // 512 threads
  unsigned n = counters[b];
  if (n > CAND_CAP) n = CAND_CAP;
  for (int i = t; i < CAND_CAP; i += 512)
    s[i] = (i < (int)n) ? cand[b * CAND_CAP + i] : 0ULL;
  __syncthreads();
  for (unsigned k2 = 2; k2 <= CAND_CAP; k2 <<= 1) {
    for (unsigned j = k2 >> 1; j > 0; j >>= 1) {
      for (int i = t; i < CAND_CAP; i += 512) {
        const unsigned ixj = (unsigned)i ^ j;
        if (ixj > (unsigned)i) {
          const unsigned long long a = s[i], c = s[ixj];
          const bool desc = (((unsigned)i & k2) == 0);
          if (desc ? (a < c) : (a > c)) { s[i] = c; s[ixj] = a; }
        }
      }
      __syncthreads();
    }
  }
  for (int j = t; j < NUM_SAMPLES; j += 512) {
    const unsigned long long c = s[j];
    const unsigned key = (unsigned)(c >> 32);
    const unsigned idx = ~(unsigned)(c & 0xFFFFFFFFull);
    tk_idx[b * NUM_SAMPLES + j] = (int)idx;
    const unsigned u = (key & 0x80000000u) ? (key & 0x7FFFFFFFu) : ~key;
    tk_sc[b * NUM_SAMPLES + j] = __uint_as_float(u);
  }
}

// ---------------- Kernel 6: gather pos/feat + sigmoid, valid masking ------
__global__ void k_gather(const float* __restrict__ feat,
                         const float* __restrict__ pos,
                         const int* __restrict__ kvals,
                         const int* __restrict__ tk_idx,
                         const float* __restrict__ tk_sc,
                         float* __restrict__ out_pos,
                         float* __restrict__ out_feat,
                         float* __restrict__ out_sc) {
  const int b = blockIdx.x / NUM_SAMPLES;
  const int j = blockIdx.x % NUM_SAMPLES;
  const int t = threadIdx.x;
  const int idx = tk_idx[b * NUM_SAMPLES + j];
  const bool valid = (j < kvals[b]) && ((unsigned)idx < (unsigned)NN);
  const size_t src = (size_t)b * NN + (unsigned)(valid ? idx : 0);

  out_feat[((size_t)b * NUM_SAMPLES + j) * DD + t] = valid ? feat[src * DD + t] : 0.f;
  if (t < PP)
    out_pos[((size_t)b * NUM_SAMPLES + j) * PP + t] = valid ? pos[src * PP + t] : 0.f;
  if (t == 0) {
    const float sc = tk_sc[b * NUM_SAMPLES + j];
    out_sc[b * NUM_SAMPLES + j] = valid ? (1.f / (1.f + expf(-sc))) : 0.f;
  }
}

extern "C" void kernel_launch(void* const* d_in, const int* in_sizes, int n_in,
                              void* d_out, int out_size, void* d_ws, size_t ws_size,
                              hipStream_t stream) {
  const float* features  = (const float*)d_in[0];  // [B,N,D]
  const float* positions = (const float*)d_in[1];  // [B,N,P]
  const int*   mask      = (const int*)d_in[2];    // [B,N]
  const float* W         = (const float*)d_in[3];  // [D]
  const float* bias      = (const float*)d_in[4];  // scalar

  // Outputs concatenated flat: pos [8,500,3] | feat [8,500,256] | scores [8,500]
  float* out_pos  = (float*)d_out;
  float* out_feat = out_pos + (size_t)BB * NUM_SAMPLES * PP;
  float* out_sc   = out_feat + (size_t)BB * NUM_SAMPLES * DD;

  // Workspace layout
  char* ws = (char*)d_ws;
  float*    scores   = (float*)ws;                               // B*N floats
  int*      kvals    = (int*)(ws + (size_t)BB * NN * 4);         // B
  unsigned* counters = (unsigned*)(ws + (size_t)BB * NN * 4 + 64);
  unsigned* Tvals    = (unsigned*)(ws + (size_t)BB * NN * 4 + 128);
  unsigned long long* cand =
      (unsigned long long*)(ws + (size_t)BB * NN * 4 + 256);     // B*1024 u64
  int*   tk_idx = (int*)(ws + (size_t)BB * NN * 4 + 256 + (size_t)BB * CAND_CAP * 8);
  float* tk_sc  = (float*)(ws + (size_t)BB * NN * 4 + 256 + (size_t)BB * CAND_CAP * 8 +
                           (size_t)BB * NUM_SAMPLES * 4);

  // 1) counts -> k, zero counters
  k_count<<<BB, 256, 0, stream>>>(mask, kvals, counters);

  // 2) WMMA scores: (B*N/16) tiles, 8 waves per block
  k_scores<<<(BB * NN / 16) / 8, 256, 0, stream>>>(features, W, bias, mask, scores);

  // 3) radix select threshold per batch
  k_radix<<<BB, 256, 0, stream>>>(scores, kvals, Tvals);

  // 4) compact candidates
  {
    dim3 g(NN / 256, BB);
    k_compact<<<g, 256, 0, stream>>>(scores, Tvals, counters, cand);
  }

  // 5) sort + emit top-500
  k_sort<<<BB, 512, 0, stream>>>(cand, counters, tk_idx, tk_sc);

  // 6) gather outputs
  k_gather<<<BB * NUM_SAMPLES, 256, 0, stream>>>(features, positions, kvals,
                                                 tk_idx, tk_sc,
                                                 out_pos, out_feat, out_sc);
}